// GNN_79663053406797
// MI455X (gfx1250) — compile-verified
//
#include <hip/hip_runtime.h>

#define N_NODES  50000
#define N_EDGES  800000
#define IN_CH    128
#define HID      256
#define PROJ     128
#define N_GRAPHS 512

typedef __attribute__((ext_vector_type(2))) float v2f;
typedef __attribute__((ext_vector_type(8))) float v8f;

// ---------------------------------------------------------------------------
// Edge scatter: agg[dst[e]] += h[src[e]]  (float4 vectorized, C/4 lanes/edge)
// ---------------------------------------------------------------------------
template <int C>
__global__ __launch_bounds__(256) void scatter_add_kernel(
    const float* __restrict__ h, const int* __restrict__ src,
    const int* __restrict__ dst, float* __restrict__ agg) {
  constexpr int VP = C / 4;
  long long gid = (long long)blockIdx.x * blockDim.x + threadIdx.x;
  if (gid >= (long long)N_EDGES * VP) return;
  int e = (int)(gid / VP);
  int p = (int)(gid % VP);
  int s = src[e];
  int d = dst[e];
  float4 v = ((const float4*)(h + (size_t)s * C))[p];
  float* o = agg + (size_t)d * C + (size_t)p * 4;
  atomicAdd(o + 0, v.x);
  atomicAdd(o + 1, v.y);
  atomicAdd(o + 2, v.z);
  atomicAdd(o + 3, v.w);
}

// ---------------------------------------------------------------------------
// f32 WMMA GEMM: Out[M,N] = act( (A (+Agg)) @ W[K,N] + bias )
// Block = 16 M-rows, 256 threads = 8 waves; wave owns N/128 16x16 tiles.
// Uses V_WMMA_F32_16X16X4_F32, K stepped by 4. M, N multiples of 16 (exact).
// A-frag layout (32-bit A 16x4): lanes 0-15 -> K=0,1 ; lanes 16-31 -> K=2,3.
// C/D layout: VGPR v = row (v + 8*(lane>=16)), col = lane&15.
// ---------------------------------------------------------------------------
template <int K, int N, bool RELU>
__global__ __launch_bounds__(256) void gemm_wmma_kernel(
    const float* __restrict__ A, const float* __restrict__ Agg,
    const float* __restrict__ W, const float* __restrict__ bias,
    float* __restrict__ Out) {
  constexpr int TPW = N / 128;            // 16x16 tiles per wave (8 waves)
  __shared__ float aTile[16 * K];

  const int mBase = blockIdx.x * 16;

  // Cooperative load of (A + Agg) 16xK tile into LDS. 16*K is a multiple of
  // 256, so every lane iterates the same count (no divergence before WMMA).
  for (int idx = threadIdx.x; idx < 16 * K; idx += 256) {
    int row = idx / K;
    int col = idx - row * K;
    size_t g = (size_t)(mBase + row) * K + col;
    float v = A[g];
    if (Agg) v += Agg[g];
    aTile[idx] = v;
  }
  __syncthreads();

  const int lane = threadIdx.x & 31;
  const int wave = threadIdx.x >> 5;
  const int r    = lane & 15;             // row (A) / col (B,C)
  const int kk   = (lane >> 4) << 1;      // 0 for lanes 0-15, 2 for 16-31

  for (int t = 0; t < TPW; ++t) {
    const int nBase = (wave * TPW + t) * 16;
    v8f c = {};
#pragma unroll 8
    for (int k0 = 0; k0 < K; k0 += 4) {
      v2f a;
      a.x = aTile[r * K + k0 + kk];
      a.y = aTile[r * K + k0 + kk + 1];
      v2f b;
      b.x = W[(size_t)(k0 + kk) * N + nBase + r];
      b.y = W[(size_t)(k0 + kk + 1) * N + nBase + r];
      // 8 args: (neg_a, A, neg_b, B, c_mod, C, reuse_a, reuse_b)
      c = __builtin_amdgcn_wmma_f32_16x16x4_f32(
          false, a, false, b, (short)0, c, false, false);
    }
    const float bi = bias[nBase + r];
    const int rowOff = (lane < 16) ? 0 : 8;
#pragma unroll
    for (int v = 0; v < 8; ++v) {
      float val = c[v] + bi;
      if (RELU) val = fmaxf(val, 0.0f);
      Out[(size_t)(mBase + rowOff + v) * N + nBase + r] = val;
    }
  }
}

// ---------------------------------------------------------------------------
// Mean-pool part 1: per-graph sums + counts via atomics (64 lanes per node).
// ---------------------------------------------------------------------------
__global__ __launch_bounds__(256) void pool_kernel(
    const float* __restrict__ h2, const int* __restrict__ batch,
    float* __restrict__ gsum, float* __restrict__ gcnt) {
  constexpr int VP = HID / 4;  // 64
  long long gid = (long long)blockIdx.x * blockDim.x + threadIdx.x;
  if (gid >= (long long)N_NODES * VP) return;
  int node = (int)(gid / VP);
  int p    = (int)(gid % VP);
  int g = batch[node];
  float4 v = ((const float4*)(h2 + (size_t)node * HID))[p];
  float* o = gsum + (size_t)g * HID + (size_t)p * 4;
  atomicAdd(o + 0, v.x);
  atomicAdd(o + 1, v.y);
  atomicAdd(o + 2, v.z);
  atomicAdd(o + 3, v.w);
  if (p == 0) atomicAdd(&gcnt[g], 1.0f);
}

// Mean-pool part 2: divide in place by max(count, 1).
__global__ __launch_bounds__(256) void divide_kernel(
    float* __restrict__ gsum, const float* __restrict__ gcnt) {
  int i = blockIdx.x * blockDim.x + threadIdx.x;
  if (i >= N_GRAPHS * HID) return;
  float c = gcnt[i / HID];
  gsum[i] = gsum[i] / fmaxf(c, 1.0f);
}

// ---------------------------------------------------------------------------
// Orchestration
// ---------------------------------------------------------------------------
extern "C" void kernel_launch(void* const* d_in, const int* in_sizes, int n_in,
                              void* d_out, int out_size, void* d_ws, size_t ws_size,
                              hipStream_t stream) {
  (void)in_sizes; (void)n_in; (void)out_size; (void)ws_size;

  const float* x    = (const float*)d_in[0];
  const int*   ei   = (const int*)d_in[1];     // [2, N_EDGES] int32
  const int*   batch= (const int*)d_in[2];
  const float* W1   = (const float*)d_in[3];
  const float* b1   = (const float*)d_in[4];
  const float* W2   = (const float*)d_in[5];
  const float* b2   = (const float*)d_in[6];
  const float* P1   = (const float*)d_in[7];
  const float* pb1  = (const float*)d_in[8];
  const float* P2   = (const float*)d_in[9];
  const float* pb2  = (const float*)d_in[10];

  const int* src = ei;
  const int* dst = ei + N_EDGES;

  // Workspace carve-out (bytes):
  //   agg : N_NODES*HID*4 = 51,200,000 (reused for both layers)
  //   h1  : N_NODES*HID*4
  //   h2  : N_NODES*HID*4
  //   gsum: N_GRAPHS*HID*4, gcnt: N_GRAPHS*4, tbuf: N_GRAPHS*HID*4
  char* ws = (char*)d_ws;
  const size_t BIG = (size_t)N_NODES * HID * sizeof(float);
  float* agg  = (float*)(ws);
  float* h1   = (float*)(ws + BIG);
  float* h2   = (float*)(ws + 2 * BIG);
  float* gsum = (float*)(ws + 3 * BIG);
  float* gcnt = gsum + (size_t)N_GRAPHS * HID;
  float* tbuf = gcnt + N_GRAPHS;

  // ---- Layer 1: agg = scatter(x); h1 = relu((x+agg) @ W1 + b1) ----
  hipMemsetAsync(agg, 0, (size_t)N_NODES * IN_CH * sizeof(float), stream);
  {
    long long tot = (long long)N_EDGES * (IN_CH / 4);
    scatter_add_kernel<IN_CH><<<(int)((tot + 255) / 256), 256, 0, stream>>>(
        x, src, dst, agg);
  }
  gemm_wmma_kernel<IN_CH, HID, true><<<N_NODES / 16, 256, 0, stream>>>(
      x, agg, W1, b1, h1);

  // ---- Layer 2: agg = scatter(h1); h2 = relu((h1+agg) @ W2 + b2) ----
  hipMemsetAsync(agg, 0, (size_t)N_NODES * HID * sizeof(float), stream);
  {
    long long tot = (long long)N_EDGES * (HID / 4);
    scatter_add_kernel<HID><<<(int)((tot + 255) / 256), 256, 0, stream>>>(
        h1, src, dst, agg);
  }
  gemm_wmma_kernel<HID, HID, true><<<N_NODES / 16, 256, 0, stream>>>(
      h1, agg, W2, b2, h2);

  // ---- Global mean pool ----
  hipMemsetAsync(gsum, 0, ((size_t)N_GRAPHS * HID + N_GRAPHS) * sizeof(float),
                 stream);
  {
    long long tot = (long long)N_NODES * (HID / 4);
    pool_kernel<<<(int)((tot + 255) / 256), 256, 0, stream>>>(
        h2, batch, gsum, gcnt);
  }
  divide_kernel<<<(N_GRAPHS * HID + 255) / 256, 256, 0, stream>>>(gsum, gcnt);

  // ---- Projection head ----
  gemm_wmma_kernel<HID, HID, true><<<N_GRAPHS / 16, 256, 0, stream>>>(
      gsum, nullptr, P1, pb1, tbuf);
  gemm_wmma_kernel<HID, PROJ, false><<<N_GRAPHS / 16, 256, 0, stream>>>(
      tbuf, nullptr, P2, pb2, (float*)d_out);
}